// LiquidDefault_90263032693032
// MI455X (gfx1250) — compile-verified
//
#include <hip/hip_runtime.h>

// ---------------------------------------------------------------------------
// LTC (liquid time-constant) 2-layer RNN, fused scan, gfx1250 (MI455X).
//   x: (256,256,8) f32 -> out: (256,1) f32
// Layer1: in=8, units=64, out=32; Layer2: in=32, units=32, out=16; FC: 16->1
// ---------------------------------------------------------------------------

#define BATCH   256
#define TSTEPS  256
#define IN1     8
#define U1      64
#define U2      32
#define OUT2    16
#define UNFOLDS 6
#define TB      2            // batch elements per block
#define NTHR    (U1 * TB)    // 128 threads = 4 waves (wave32)

typedef __attribute__((ext_vector_type(2))) float v2f;
typedef __attribute__((ext_vector_type(8))) float v8f;

__device__ __forceinline__ float fast_sigmoid(float t) {
    // 1/(1+exp(-t)) = 1/(1+exp2(-t*log2(e)))  -> v_exp_f32 + v_rcp_f32
    float e = __builtin_amdgcn_exp2f(t * -1.4426950408889634f);
    return __builtin_amdgcn_rcpf(1.0f + e);
}

// ------------------------- parameter pre-fold kernel -----------------------
struct PrepArgs {
    const float *sigma1, *mu1, *w1, *erev1, *mask1;
    const float *ss1, *sm1, *sw1, *serev1, *smask1;
    const float *gleak1, *vleak1, *cm1;
    const float *iw1, *ib1, *ow1, *ob1;
    const float *sigma2, *mu2, *w2, *erev2, *mask2;
    const float *ss2, *sm2, *sw2, *serev2, *smask2;
    const float *gleak2, *vleak2, *cm2;
    const float *iw2, *ib2, *ow2, *ob2;
    const float *fcw, *fcb;
};

__global__ void ltc_prep(PrepArgs a, float4* rec1, float4* sen1,
                         float4* rec2, float4* sen2,
                         float4* unit1, float4* unit2, float* misc) {
    const int tid = threadIdx.x, n = blockDim.x;
    for (int idx = tid; idx < U1 * U1; idx += n) {
        float wm = a.w1[idx] * a.mask1[idx];
        rec1[idx] = make_float4(a.sigma1[idx], a.mu1[idx], wm, wm * a.erev1[idx]);
    }
    for (int idx = tid; idx < IN1 * U1; idx += n) {
        float wm = a.sw1[idx] * a.smask1[idx];
        sen1[idx] = make_float4(a.ss1[idx], a.sm1[idx], wm, wm * a.serev1[idx]);
    }
    for (int idx = tid; idx < U2 * U2; idx += n) {
        float wm = a.w2[idx] * a.mask2[idx];
        rec2[idx] = make_float4(a.sigma2[idx], a.mu2[idx], wm, wm * a.erev2[idx]);
    }
    for (int idx = tid; idx < U2 * U2; idx += n) {
        float wm = a.sw2[idx] * a.smask2[idx];
        sen2[idx] = make_float4(a.ss2[idx], a.sm2[idx], wm, wm * a.serev2[idx]);
    }
    for (int idx = tid; idx < U1; idx += n)
        unit1[idx] = make_float4(a.cm1[idx] * (float)UNFOLDS,
                                 a.gleak1[idx] * a.vleak1[idx], a.gleak1[idx], 0.f);
    for (int idx = tid; idx < U2; idx += n)
        unit2[idx] = make_float4(a.cm2[idx] * (float)UNFOLDS,
                                 a.gleak2[idx] * a.vleak2[idx], a.gleak2[idx], 0.f);
    // misc: [0..7]=iw1 [8..15]=ib1 [16..47]=a2 [48..79]=b2 [80..95]=fw [96]=fb
    if (tid < IN1) { misc[tid] = a.iw1[tid]; misc[8 + tid] = a.ib1[tid]; }
    if (tid < U2) {
        misc[16 + tid] = a.ow1[tid] * a.iw2[tid];                 // fold L1-out affine
        misc[48 + tid] = a.ob1[tid] * a.iw2[tid] + a.ib2[tid];    // into L2-in affine
    }
    if (tid < OUT2) misc[80 + tid] = a.ow2[tid] * a.fcw[tid];     // fold L2-out into fc
    if (tid == 0) {
        float s = a.fcb[0];
        for (int k = 0; k < OUT2; ++k) s += a.ob2[k] * a.fcw[k];
        misc[96] = s;
    }
}

// ------------------------- fused two-layer scan ----------------------------
__global__ __launch_bounds__(NTHR)
void ltc_main(const float* __restrict__ x,
              const float4* __restrict__ gRec1, const float4* __restrict__ gSen1,
              const float4* __restrict__ gRec2, const float4* __restrict__ gSen2,
              const float4* __restrict__ gUnit1, const float4* __restrict__ gUnit2,
              const float* __restrict__ gMisc, float* __restrict__ y16) {
    __shared__ float4 sRec1[U1 * U1];       // 64 KB
    __shared__ float4 sSen1[IN1 * U1];      //  8 KB
    __shared__ float4 sRec2[U2 * U2];       // 16 KB
    __shared__ float4 sSen2[U2 * U2];       // 16 KB
    __shared__ float4 sUnit1[U1];
    __shared__ float4 sUnit2[U2];
    __shared__ float  sIw1[IN1], sIb1[IN1], sA2[U2], sB2[U2], sFw[OUT2];
    __shared__ float  v1buf[2][TB][U1];
    __shared__ float  v2buf[2][TB][U2];
    __shared__ float  sXt[TB][IN1];

    const int j   = threadIdx.x;               // unit index (0..63)
    const int ty  = threadIdx.y;               // local batch (0..TB-1)
    const int tid = ty * U1 + j;
    const int bg  = blockIdx.x * TB + ty;      // global batch

    // cooperative parameter staging to LDS (reused 256*6 times)
    for (int idx = tid; idx < U1 * U1; idx += NTHR) sRec1[idx] = gRec1[idx];
    for (int idx = tid; idx < IN1 * U1; idx += NTHR) sSen1[idx] = gSen1[idx];
    for (int idx = tid; idx < U2 * U2; idx += NTHR) sRec2[idx] = gRec2[idx];
    for (int idx = tid; idx < U2 * U2; idx += NTHR) sSen2[idx] = gSen2[idx];
    for (int idx = tid; idx < U1; idx += NTHR) sUnit1[idx] = gUnit1[idx];
    for (int idx = tid; idx < U2; idx += NTHR) sUnit2[idx] = gUnit2[idx];
    if (tid < IN1) { sIw1[tid] = gMisc[tid]; sIb1[tid] = gMisc[8 + tid]; }
    if (tid < U2)  { sA2[tid] = gMisc[16 + tid]; sB2[tid] = gMisc[48 + tid]; }
    if (tid < OUT2) sFw[tid] = gMisc[80 + tid];

    v1buf[0][ty][j] = 0.0f;
    if (j < U2) v2buf[0][ty][j] = 0.0f;
    __syncthreads();

    const float4 u1 = sUnit1[j];
    const float4 u2 = (j < U2) ? sUnit2[j] : make_float4(0.f, 0.f, 0.f, 0.f);

    int cur1 = 0, cur2 = 0;
    for (int t = 0; t < TSTEPS; ++t) {
        // stage & scale the input sample for this step
        if (j < IN1)
            sXt[ty][j] = x[(bg * TSTEPS + t) * IN1 + j] * sIw1[j] + sIb1[j];
        __syncthreads();

        // ---- layer 1 sensory (once per step) ----
        float wns1 = 0.f, wds1 = 0.f;
        #pragma unroll
        for (int k = 0; k < IN1; ++k) {
            float4 p = sSen1[k * U1 + j];
            float s = fast_sigmoid(p.x * (sXt[ty][k] - p.y));
            wds1 += p.z * s;
            wns1 += p.w * s;
        }
        // ---- layer 1 ODE unfolds ----
        for (int u = 0; u < UNFOLDS; ++u) {
            float num = 0.f, den = 0.f;
            #pragma unroll 8
            for (int i = 0; i < U1; ++i) {
                float4 p = sRec1[i * U1 + j];
                float s = fast_sigmoid(p.x * (v1buf[cur1][ty][i] - p.y));
                den += p.z * s;
                num += p.w * s;
            }
            float vj = v1buf[cur1][ty][j];
            float vn = (u1.x * vj + u1.y + num + wns1) *
                       __builtin_amdgcn_rcpf(u1.x + u1.z + den + wds1 + 1e-8f);
            v1buf[cur1 ^ 1][ty][j] = vn;
            __syncthreads();
            cur1 ^= 1;
        }

        // ---- layer 2 sensory: consumes layer 1 state of this step ----
        float wns2 = 0.f, wds2 = 0.f;
        if (j < U2) {
            #pragma unroll 8
            for (int k = 0; k < U2; ++k) {
                float xk = v1buf[cur1][ty][k] * sA2[k] + sB2[k];
                float4 p = sSen2[k * U2 + j];
                float s = fast_sigmoid(p.x * (xk - p.y));
                wds2 += p.z * s;
                wns2 += p.w * s;
            }
        }
        // ---- layer 2 ODE unfolds ----
        for (int u = 0; u < UNFOLDS; ++u) {
            if (j < U2) {
                float num = 0.f, den = 0.f;
                #pragma unroll 8
                for (int i = 0; i < U2; ++i) {
                    float4 p = sRec2[i * U2 + j];
                    float s = fast_sigmoid(p.x * (v2buf[cur2][ty][i] - p.y));
                    den += p.z * s;
                    num += p.w * s;
                }
                float vj = v2buf[cur2][ty][j];
                float vn = (u2.x * vj + u2.y + num + wns2) *
                           __builtin_amdgcn_rcpf(u2.x + u2.z + den + wds2 + 1e-8f);
                v2buf[cur2 ^ 1][ty][j] = vn;
            }
            __syncthreads();
            cur2 ^= 1;
        }
    }

    // emit fc-scaled last hidden state; WMMA kernel does the row-sum + sigmoid
    if (j < OUT2)
        y16[bg * OUT2 + j] = v2buf[cur2][ty][j] * sFw[j];
}

// ------------------- final head: row-sum via V_WMMA_F32_16X16X4_F32 --------
// One wave handles 16 batch rows. D = sum_k A(16x16) * ones == per-row sums
// replicated across N; chain 4 K=4 WMMAs. A 16x4 f32 layout: lane m=L&15,
// hi=L>>4 holds {A[m][2hi], A[m][2hi+1]}. D: vgpr r, lanes0-15 -> M=r,
// lanes16-31 -> M=8+r.
__global__ void ltc_fc_wmma(const float* __restrict__ y16,
                            const float* __restrict__ fbp,
                            float* __restrict__ out) {
    const int tile = blockIdx.x;           // 16 tiles of 16 rows
    const int lane = threadIdx.x;          // 0..31
    const int m  = lane & 15;
    const int hi = lane >> 4;

    v8f c = {};
    v2f ones; ones[0] = 1.0f; ones[1] = 1.0f;
    const int row = tile * 16 + m;
    #pragma unroll
    for (int kc = 0; kc < 4; ++kc) {
        v2f a;
        a[0] = y16[row * OUT2 + kc * 4 + 2 * hi + 0];
        a[1] = y16[row * OUT2 + kc * 4 + 2 * hi + 1];
        c = __builtin_amdgcn_wmma_f32_16x16x4_f32(false, a, false, ones,
                                                  (short)0, c, false, false);
    }
    const float fb = *fbp;
    if (m == 0) {  // lanes 0 and 16 hold rows 0..7 / 8..15 across vgprs
        #pragma unroll
        for (int r = 0; r < 8; ++r) {
            float t = c[r] + fb;
            out[tile * 16 + hi * 8 + r] = fast_sigmoid(t);
        }
    }
}

// ---------------------------------------------------------------------------
extern "C" void kernel_launch(void* const* d_in, const int* in_sizes, int n_in,
                              void* d_out, int out_size, void* d_ws, size_t ws_size,
                              hipStream_t stream) {
    (void)in_sizes; (void)n_in; (void)out_size; (void)ws_size;
    // setup_inputs() recursive dict-insertion-order flattening:
    const float* x = (const float*)d_in[0];
    // p1: gleak vleak cm sigma mu w erev s_sigma s_mu s_w s_erev in_w in_b out_w out_b
    const float *g1 = (const float*)d_in[1],  *vl1 = (const float*)d_in[2],
                *cm1 = (const float*)d_in[3], *sg1 = (const float*)d_in[4],
                *mu1 = (const float*)d_in[5], *w1  = (const float*)d_in[6],
                *er1 = (const float*)d_in[7], *ssg1 = (const float*)d_in[8],
                *smu1 = (const float*)d_in[9], *sw1 = (const float*)d_in[10],
                *ser1 = (const float*)d_in[11], *iw1 = (const float*)d_in[12],
                *ib1 = (const float*)d_in[13], *ow1 = (const float*)d_in[14],
                *ob1 = (const float*)d_in[15];
    const float *g2 = (const float*)d_in[16], *vl2 = (const float*)d_in[17],
                *cm2 = (const float*)d_in[18], *sg2 = (const float*)d_in[19],
                *mu2 = (const float*)d_in[20], *w2  = (const float*)d_in[21],
                *er2 = (const float*)d_in[22], *ssg2 = (const float*)d_in[23],
                *smu2 = (const float*)d_in[24], *sw2 = (const float*)d_in[25],
                *ser2 = (const float*)d_in[26], *iw2 = (const float*)d_in[27],
                *ib2 = (const float*)d_in[28], *ow2 = (const float*)d_in[29],
                *ob2 = (const float*)d_in[30];
    const float *mask1 = (const float*)d_in[31], *smask1 = (const float*)d_in[32],
                *mask2 = (const float*)d_in[33], *smask2 = (const float*)d_in[34],
                *fcw = (const float*)d_in[35], *fcb = (const float*)d_in[36];

    // workspace layout
    float4* wRec1 = (float4*)d_ws;              // 4096
    float4* wSen1 = wRec1 + U1 * U1;            // 512
    float4* wRec2 = wSen1 + IN1 * U1;           // 1024
    float4* wSen2 = wRec2 + U2 * U2;            // 1024
    float4* wUnit1 = wSen2 + U2 * U2;           // 64
    float4* wUnit2 = wUnit1 + U1;               // 32
    float*  wMisc = (float*)(wUnit2 + U2);      // 128 floats
    float*  wY16  = wMisc + 128;                // 256*16 floats

    PrepArgs pa{sg1, mu1, w1, er1, mask1, ssg1, smu1, sw1, ser1, smask1,
                g1, vl1, cm1, iw1, ib1, ow1, ob1,
                sg2, mu2, w2, er2, mask2, ssg2, smu2, sw2, ser2, smask2,
                g2, vl2, cm2, iw2, ib2, ow2, ob2, fcw, fcb};

    ltc_prep<<<1, 256, 0, stream>>>(pa, wRec1, wSen1, wRec2, wSen2,
                                    wUnit1, wUnit2, wMisc);

    dim3 blk(U1, TB);
    ltc_main<<<BATCH / TB, blk, 0, stream>>>(x, wRec1, wSen1, wRec2, wSen2,
                                             wUnit1, wUnit2, wMisc, wY16);

    ltc_fc_wmma<<<BATCH / 16, 32, 0, stream>>>(wY16, wMisc + 96, (float*)d_out);
}